// MTL_CE_51977694216656
// MI455X (gfx1250) — compile-verified
//
#include <hip/hip_runtime.h>
#include <hip/hip_bf16.h>

// Grouped (per-task) GEMV: pred[n] = dot(X[n,:], W[:, task_ids[n]])
// N ~ 1e6, D = 256, T = 64, fp32. Memory-bound: stream X once, W lives in LDS.
// Uses V_WMMA_F32_16X16X4_F32: A = 16 rows of X (K-slice), B = gathered weight
// columns per row; the diagonal of the 16x16 accumulator is the answer.

typedef float v2f __attribute__((ext_vector_type(2)));
typedef float v8f __attribute__((ext_vector_type(8)));

#define DDIM 256
#define TASKS 64
#define LDS_STRIDE 258   // even (8B-aligned b64 LDS loads) + bank spread (258%64==2)

__global__ __launch_bounds__(256)
void MTL_CE_grouped_gemv_wmma(const float* __restrict__ X,
                              const float* __restrict__ W,
                              const int*   __restrict__ tids,
                              float* __restrict__ out,
                              int n, int nChunks, int totalWaves)
{
    __shared__ float sW[TASKS * LDS_STRIDE];   // Wt[col][k], padded rows

    const int tid = threadIdx.x;

    // Stage W (64KB) into LDS transposed: W is row-major [D=256][T=64].
    for (int i = tid; i < DDIM * TASKS; i += 256) {
        const int k   = i >> 6;          // D index
        const int col = i & (TASKS - 1); // task index
        sW[col * LDS_STRIDE + k] = W[i];
    }
    __syncthreads();

    const int lane = tid & 31;
    const int half = lane >> 4;   // 0: K = k0..k0+1 ; 1: K = k0+2..k0+3
    const int lrow = lane & 15;   // A-matrix row / B-matrix column index
    const int wave = blockIdx.x * (256 >> 5) + (tid >> 5);

    for (int chunk = wave; chunk < nChunks; chunk += totalWaves) {
        const int r0   = chunk * 16;
        int       row  = r0 + lrow;
        const int rowc = (row < n) ? row : (n - 1);        // tail clamp (loads only)
        const int col  = tids[rowc] & (TASKS - 1);

        const float* xrow = X  + (size_t)rowc * DDIM + half * 2;
        const float* wcol = sW + col * LDS_STRIDE    + half * 2;

        v8f c = {};
        #pragma unroll 8
        for (int k0 = 0; k0 < DDIM; k0 += 4) {
            v2f a = *(const v2f*)(xrow + k0);   // global_load_b64, A 16x4 f32 layout
            v2f b = *(const v2f*)(wcol + k0);   // ds_load_b64,    B 4x16 f32 layout
            // D = A*B + C   (f32, round-to-nearest-even)
            c = __builtin_amdgcn_wmma_f32_16x16x4_f32(
                    /*neg_a=*/false, a, /*neg_b=*/false, b,
                    /*c_mod=*/(short)0, c, /*reuse_a=*/false, /*reuse_b=*/false);
        }

        // Diagonal extraction from the 16x16 f32 C/D layout:
        //   m in 0..7  -> lane m      , VGPR m
        //   m in 8..15 -> lane 16 + m , VGPR m-8
        const int idx = (lane < 8) ? lane : (lane - 24);
        float r = c[0];
        r = (idx == 1) ? c[1] : r;
        r = (idx == 2) ? c[2] : r;
        r = (idx == 3) ? c[3] : r;
        r = (idx == 4) ? c[4] : r;
        r = (idx == 5) ? c[5] : r;
        r = (idx == 6) ? c[6] : r;
        r = (idx == 7) ? c[7] : r;

        const bool active = (lane < 8) || (lane >= 24);
        const int  orow   = r0 + ((lane < 8) ? lane : (lane - 16));
        if (active && orow < n) out[orow] = r;
    }
}

extern "C" void kernel_launch(void* const* d_in, const int* in_sizes, int n_in,
                              void* d_out, int out_size, void* d_ws, size_t ws_size,
                              hipStream_t stream) {
    const float* X  = (const float*)d_in[0];   // [N, 256] fp32
    const float* W  = (const float*)d_in[1];   // [256, 64] fp32
    const int*   t  = (const int*)d_in[2];     // [N] task ids (sorted)
    float*       o  = (float*)d_out;           // [N] fp32

    const int n       = in_sizes[2];
    const int nChunks = (n + 15) / 16;

    const int wavesPerBlock = 256 / 32;
    int blocks = 2048;
    if (blocks * wavesPerBlock > nChunks)
        blocks = (nChunks + wavesPerBlock - 1) / wavesPerBlock;
    if (blocks < 1) blocks = 1;
    const int totalWaves = blocks * wavesPerBlock;

    MTL_CE_grouped_gemv_wmma<<<blocks, 256, 0, stream>>>(X, W, t, o, n, nChunks, totalWaves);
}